// UpdatePositionLayer_75565654606297
// MI455X (gfx1250) — compile-verified
//
#include <hip/hip_runtime.h>
#include <math.h>

typedef float v2f __attribute__((ext_vector_type(2)));
typedef float v8f __attribute__((ext_vector_type(8)));

#define TILE_E   64      // edges per block
#define BLOCK_T  128     // 4 waves (wave32)
#define HSTRIDE  260     // 256 + pad (bank-conflict-free A reads)
#define WSTR1    273     // 272 + pad
#define WSTR2    68      // 64 + pad
#define NCOL1    272
#define NCOL2    64
#define NCT1     17
#define NCT2     4

#define INV_NN     0.5129891760425771f   // 1/sqrt(3.8)
#define INV_SQRT3  0.5773502691896258f
#define SQRT3F     1.7320508075688772f
#define INV16      0.0625f               // 1/sqrt(256)
#define INV_SQRT6  0.4082482904638630f

// ------------------- CG tables: exact device-side port of reference ---------
__device__ double dfact(int n){ double r=1.0; for(int i=2;i<=n;++i) r*=(double)i; return r; }

__device__ double cg_coef_dev(int j1,int m1,int j2,int m2,int j3,int m3){
  double pref = sqrt((double)(2*j3+1)*dfact(j3+j1-j2)*dfact(j3-j1+j2)*dfact(j1+j2-j3)/dfact(j1+j2+j3+1));
  pref *= sqrt(dfact(j3+m3)*dfact(j3-m3)*dfact(j1-m1)*dfact(j1+m1)*dfact(j2-m2)*dfact(j2+m2));
  double s = 0.0;
  for(int k=0;k<=j1+j2-j3;++k){
    int a0=k, a1=j1+j2-j3-k, a2=j1-m1-k, a3=j2+m2-k, a4=j3-j2+m1+k, a5=j3-j1-m2+k;
    if(a0<0||a1<0||a2<0||a3<0||a4<0||a5<0) continue;
    double d = dfact(a0)*dfact(a1)*dfact(a2)*dfact(a3)*dfact(a4)*dfact(a5);
    s += ((k&1)? -1.0:1.0)/d;
  }
  return pref*s;
}

__device__ void buildU(int l, double Ur[7][7], double Ui[7][7]){
  for(int i=0;i<7;++i) for(int j=0;j<7;++j){ Ur[i][j]=0.0; Ui[i][j]=0.0; }
  const double is2 = 0.7071067811865476;
  Ur[l][l] = 1.0;
  for(int m=1;m<=l;++m){
    double sg = (m&1)? -1.0 : 1.0;
    Ur[l+m][l+m] = sg*is2;
    Ur[l+m][l-m] = is2;
    Ui[l-m][l-m] = is2;
    Ui[l-m][l+m] = -sg*is2;
  }
}

// slots: 0:(0,0,0) 1:(1,1,0) 2:(2,2,0) 3:(3,3,0) 4:(0,1,1) 5:(1,0,1)
//        6:(1,2,1) 7:(2,1,1) 8:(2,3,1) 9:(3,2,1) 10:(1,1,1) 11:(2,2,1) 12:(3,3,1)
__global__ void cg_init_kernel(float* __restrict__ cg){
  int t = blockIdx.x*blockDim.x + threadIdx.x;
  if (t >= 13) return;
  const int L1s[13]={0,1,2,3,0,1,1,2,2,3,1,2,3};
  const int L2s[13]={0,1,2,3,1,0,2,1,3,2,1,2,3};
  const int L3s[13]={0,0,0,0,1,1,1,1,1,1,1,1,1};
  int l1=L1s[t], l2=L2s[t], l3=L3s[t];
  int n1=2*l1+1, n2=2*l2+1, n3=2*l3+1;
  double CGc[7][7][7];
  for(int a=0;a<n1;++a)for(int b=0;b<n2;++b)for(int c=0;c<n3;++c) CGc[a][b][c]=0.0;
  for(int m1=-l1;m1<=l1;++m1)
    for(int m2=-l2;m2<=l2;++m2){
      int m3=m1+m2;
      if(m3>=-l3 && m3<=l3) CGc[m1+l1][m2+l2][m3+l3]=cg_coef_dev(l1,m1,l2,m2,l3,m3);
    }
  double U1r[7][7],U1i[7][7],U2r[7][7],U2i[7][7],U3r[7][7],U3i[7][7];
  buildU(l1,U1r,U1i); buildU(l2,U2r,U2i); buildU(l3,U3r,U3i);
  double Cre[7][7][7], Cim[7][7][7];
  double sre=0.0, sim=0.0;
  for(int a=0;a<n1;++a)for(int b=0;b<n2;++b)for(int c=0;c<n3;++c){
    double accr=0.0, acci=0.0;
    for(int u=0;u<n1;++u){
      double ur=U1r[a][u], ui=U1i[a][u];
      if(ur==0.0 && ui==0.0) continue;
      for(int v=0;v<n2;++v){
        double vr=U2r[b][v], vi=U2i[b][v];
        if(vr==0.0 && vi==0.0) continue;
        double pr=ur*vr-ui*vi, pi=ur*vi+ui*vr;
        for(int w=0;w<n3;++w){
          double g=CGc[u][v][w];
          if(g==0.0) continue;
          double wr=U3r[c][w], wi=-U3i[c][w];   // conj
          accr += (pr*wr-pi*wi)*g;
          acci += (pr*wi+pi*wr)*g;
        }
      }
    }
    Cre[a][b][c]=accr; Cim[a][b][c]=acci;
    sre += fabs(accr); sim += fabs(acci);
  }
  bool useIm = sim > sre;
  float* dst = cg + t*256;
  for(int a=0;a<n1;++a)for(int b=0;b<n2;++b)for(int c=0;c<n3;++c)
    dst[(a*n2+b)*n3+c] = (float)(useIm? Cim[a][b][c] : Cre[a][b][c]);
}

// --------------------- edge geometry: sph harm + radial basis ---------------
__global__ void edge_geom_kernel(const float* __restrict__ pos,
                                 const int* __restrict__ esrc,
                                 const int* __restrict__ edst,
                                 float* __restrict__ ea,
                                 float* __restrict__ es,
                                 float* __restrict__ x0, int E){
  int e = blockIdx.x*blockDim.x + threadIdx.x;
  if (e >= E) return;
  int s = esrc[e], d = edst[e];
  float ex = pos[s*3+0]-pos[d*3+0];
  float ey = pos[s*3+1]-pos[d*3+1];
  float ez = pos[s*3+2]-pos[d*3+2];
  float r  = sqrtf(ex*ex+ey*ey+ez*ez);
  float ri = 1.0f/r;
  float x = ex*ri, y = ey*ri, z = ez*ri;
  const float s3=1.7320508075688772f, s5=2.2360679774997896f, s15=3.8729833462074170f;
  const float c33=2.0916500663351889f, c32=10.246950765959598f, c31=1.6201851746019651f;
  const float c30=1.3228756555322954f, c3p2=5.1234753829797990f;
  float c[16];
  c[0]=1.0f; c[1]=s3*y; c[2]=s3*z; c[3]=s3*x;
  c[4]=s15*x*y; c[5]=s15*y*z; c[6]=0.5f*s5*(3.0f*z*z-1.0f); c[7]=s15*x*z; c[8]=0.5f*s15*(x*x-y*y);
  c[9]=c33*y*(3.0f*x*x-y*y); c[10]=c32*x*y*z; c[11]=c31*y*(5.0f*z*z-1.0f);
  c[12]=c30*(5.0f*z*z*z-3.0f*z); c[13]=c31*x*(5.0f*z*z-1.0f);
  c[14]=c3p2*z*(x*x-y*y); c[15]=c33*x*(x*x-3.0f*y*y);
  #pragma unroll
  for(int j=0;j<16;++j){
    ea[(size_t)e*16+j]=c[j];
    atomicAdd(&x0[(size_t)d*16+j], c[j]*INV_NN);
  }
  const float cen[3]={1.0f,1.5f,2.0f};
  #pragma unroll
  for(int j=0;j<3;++j){
    float dd=(r-cen[j])*2.0f;
    float d2=dd*dd;
    float v = (d2 < 1.0f) ? 1.14136f*expf(2.0f + 1.0f/(d2-1.0f)) : 0.0f;
    es[(size_t)e*3+j]= v*SQRT3F;
  }
}

// ------------- fused: edge MLP (WMMA f32 16x16x4) + tensor product 1 --------
__global__ void __launch_bounds__(BLOCK_T)
mlp_tp1_kernel(const float* __restrict__ es, const float* __restrict__ ea,
               const float* __restrict__ x0,
               const int* __restrict__ esrc, const int* __restrict__ edst,
               const float* __restrict__ W1, const float* __restrict__ W2,
               const float* __restrict__ cg, float* __restrict__ xg, int E){
  extern __shared__ float smem[];
  float* hsh = smem;                        // TILE_E x HSTRIDE
  float* wsh = smem + TILE_E*HSTRIDE;       // 64 x WSTR1
  int tid = threadIdx.x;
  int blockBase = blockIdx.x * TILE_E;

  // Stage A: h = relu(es @ W1 / sqrt(3))   (64 x 256 into LDS)
  {
    int el = tid & 63;
    int eg = blockBase + el;
    float e0=0.f, e1=0.f, e2=0.f;
    if (eg < E){ e0=es[(size_t)eg*3+0]; e1=es[(size_t)eg*3+1]; e2=es[(size_t)eg*3+2]; }
    for(int c = (tid>>6); c < 256; c += 2){
      float hv = (e0*W1[c] + e1*W1[256+c] + e2*W1[512+c]) * INV_SQRT3;
      hsh[el*HSTRIDE + c] = fmaxf(hv, 0.0f);
    }
  }
  __syncthreads();

  // Stage B: w = h @ W2 / 16  via V_WMMA_F32_16X16X4_F32, staged to LDS
  int wave = tid >> 5, lane = tid & 31;
  int row = lane & 15, kh = lane >> 4;
  const float* hrow = &hsh[(wave*16 + row)*HSTRIDE];
  for(int ct=0; ct<NCT1; ++ct){
    v8f acc = {0.f,0.f,0.f,0.f,0.f,0.f,0.f,0.f};
    int col = ct*16 + row;
    for(int k=0;k<256;k+=4){
      int kk = k + 2*kh;                      // A: M=lane%16, K=v+2*(lane/16)
      v2f a; a.x = hrow[kk]; a.y = hrow[kk+1];
      v2f b; b.x = W2[(size_t)kk*NCOL1 + col]; b.y = W2[(size_t)(kk+1)*NCOL1 + col];
      acc = __builtin_amdgcn_wmma_f32_16x16x4_f32(false, a, false, b, (short)0, acc, false, false);
    }
    float* wbase = &wsh[wave*16*WSTR1];
    #pragma unroll
    for(int v=0;v<8;++v)                      // D: M=v+8*(lane/16), N=lane%16
      wbase[(v + 8*kh)*WSTR1 + col] = acc[v] * INV16;
  }
  __syncthreads();

  // Stage C: tensor product (IRREPS_SH x IRREPS_SH -> GATE_IN), scatter to xg
  if (lane < 16){
    int e = blockBase + wave*16 + lane;
    if (e < E){
      const float* wl = &wsh[(wave*16 + lane)*WSTR1];
      int s = esrc[e], d = edst[e];
      float a[16], b[16];
      #pragma unroll
      for(int j=0;j<16;++j){ a[j]=x0[(size_t)s*16+j]; b[j]=ea[(size_t)e*16+j]; }
      const float* t0=cg;            const float* t1=cg+1*256;
      const float* t2=cg+2*256;      const float* t3=cg+3*256;
      const float* t4=cg+4*256;      const float* t5=cg+5*256;
      const float* t6=cg+6*256;      const float* t7=cg+7*256;
      const float* t8=cg+8*256;      const float* t9=cg+9*256;
      const float* t10=cg+10*256;    const float* t11=cg+11*256;
      const float* t12=cg+12*256;
      float s00 = a[0]*b[0]*t0[0];
      float s11=0.f, s22=0.f, s33=0.f;
      for(int i=0;i<3;++i) for(int j=0;j<3;++j) s11 += a[1+i]*b[1+j]*t1[i*3+j];
      for(int i=0;i<5;++i) for(int j=0;j<5;++j) s22 += a[4+i]*b[4+j]*t2[i*5+j];
      for(int i=0;i<7;++i) for(int j=0;j<7;++j) s33 += a[9+i]*b[9+j]*t3[i*7+j];
      float pv[9][3];
      for(int p=0;p<9;++p) for(int k=0;k<3;++k) pv[p][k]=0.f;
      for(int j=0;j<3;++j) for(int k=0;k<3;++k) pv[0][k]+= a[0]*b[1+j]*t4[j*3+k];
      for(int i=0;i<3;++i) for(int k=0;k<3;++k) pv[1][k]+= a[1+i]*b[0]*t5[i*3+k];
      for(int i=0;i<3;++i) for(int j=0;j<5;++j) for(int k=0;k<3;++k) pv[2][k]+= a[1+i]*b[4+j]*t6[(i*5+j)*3+k];
      for(int i=0;i<5;++i) for(int j=0;j<3;++j) for(int k=0;k<3;++k) pv[3][k]+= a[4+i]*b[1+j]*t7[(i*3+j)*3+k];
      for(int i=0;i<5;++i) for(int j=0;j<7;++j) for(int k=0;k<3;++k) pv[4][k]+= a[4+i]*b[9+j]*t8[(i*7+j)*3+k];
      for(int i=0;i<7;++i) for(int j=0;j<5;++j) for(int k=0;k<3;++k) pv[5][k]+= a[9+i]*b[4+j]*t9[(i*5+j)*3+k];
      for(int i=0;i<3;++i) for(int j=0;j<3;++j) for(int k=0;k<3;++k) pv[6][k]+= a[1+i]*b[1+j]*t10[(i*3+j)*3+k];
      for(int i=0;i<5;++i) for(int j=0;j<5;++j) for(int k=0;k<3;++k) pv[7][k]+= a[4+i]*b[4+j]*t11[(i*5+j)*3+k];
      for(int i=0;i<7;++i) for(int j=0;j<7;++j) for(int k=0;k<3;++k) pv[8][k]+= a[9+i]*b[9+j]*t12[(i*7+j)*3+k];
      float* xr = &xg[(size_t)d*160];
      const float c05 = 0.5f*INV_NN;          // 1/sqrt(fan=4)
      #pragma unroll
      for(int c=0;c<16;++c)
        atomicAdd(&xr[c], (wl[c]*s00 + wl[64+c]*s11 + wl[144+c]*s22 + wl[224+c]*s33)*c05);
      #pragma unroll
      for(int c=0;c<8;++c){
        atomicAdd(&xr[32+c], (wl[16+c]*s00 + wl[80+c]*s11 + wl[160+c]*s22 + wl[240+c]*s33)*c05);
        atomicAdd(&xr[48+c], (wl[24+c]*s00 + wl[88+c]*s11 + wl[168+c]*s22 + wl[248+c]*s33)*c05);
      }
      const float c6 = INV_SQRT6*INV_NN;      // 1/sqrt(fan=6)
      const float c3 = INV_SQRT3*INV_NN;      // 1/sqrt(fan=3)
      for(int c=0;c<16;++c){
        #pragma unroll
        for(int k=0;k<3;++k){
          float vo = wl[32+c]*pv[0][k] + wl[48+c]*pv[1][k] + wl[112+c]*pv[2][k]
                   + wl[128+c]*pv[3][k] + wl[192+c]*pv[4][k] + wl[208+c]*pv[5][k];
          atomicAdd(&xr[64+3*c+k], vo*c6);
          float ve = wl[96+c]*pv[6][k] + wl[176+c]*pv[7][k] + wl[256+c]*pv[8][k];
          atomicAdd(&xr[112+3*c+k], ve*c3);
        }
      }
    }
  }
}

// ----------------------------- gate nonlinearity ----------------------------
__global__ void gate_kernel(const float* __restrict__ xg, float* __restrict__ z, int N){
  int idx = blockIdx.x*blockDim.x + threadIdx.x;
  if (idx >= N*128) return;
  int n = idx >> 7, c = idx & 127;
  const float* x = &xg[(size_t)n*160];
  float out;
  if (c < 16)       out = fmaxf(x[c], 0.0f);
  else if (c < 32)  out = fabsf(x[c]);
  else if (c < 80){ int t=c-32; int i=t/3;
    float g = (i<8) ? fmaxf(x[32+i],0.0f) : tanhf(x[40+(i-8)]);
    out = x[64+t]*g;
  } else {          int t=c-80; int i=t/3;
    float g = (i<8) ? fmaxf(x[48+i],0.0f) : tanhf(x[56+(i-8)]);
    out = x[112+t]*g;
  }
  z[(size_t)n*128 + c] = out;
}

// ------------- fused: final edge MLP (WMMA) + tensor product 2 --------------
__global__ void __launch_bounds__(BLOCK_T)
mlp_tp2_kernel(const float* __restrict__ es, const float* __restrict__ ea,
               const float* __restrict__ z,
               const int* __restrict__ esrc, const int* __restrict__ edst,
               const float* __restrict__ W1f, const float* __restrict__ W2f,
               const float* __restrict__ cg, float* __restrict__ outp, int E){
  extern __shared__ float smem[];
  float* hsh = smem;
  float* wsh = smem + TILE_E*HSTRIDE;       // 64 x WSTR2
  int tid = threadIdx.x;
  int blockBase = blockIdx.x * TILE_E;
  {
    int el = tid & 63;
    int eg = blockBase + el;
    float e0=0.f, e1=0.f, e2=0.f;
    if (eg < E){ e0=es[(size_t)eg*3+0]; e1=es[(size_t)eg*3+1]; e2=es[(size_t)eg*3+2]; }
    for(int c = (tid>>6); c < 256; c += 2){
      float hv = (e0*W1f[c] + e1*W1f[256+c] + e2*W1f[512+c]) * INV_SQRT3;
      hsh[el*HSTRIDE + c] = fmaxf(hv, 0.0f);
    }
  }
  __syncthreads();
  int wave = tid >> 5, lane = tid & 31;
  int row = lane & 15, kh = lane >> 4;
  const float* hrow = &hsh[(wave*16 + row)*HSTRIDE];
  for(int ct=0; ct<NCT2; ++ct){
    v8f acc = {0.f,0.f,0.f,0.f,0.f,0.f,0.f,0.f};
    int col = ct*16 + row;
    for(int k=0;k<256;k+=4){
      int kk = k + 2*kh;
      v2f a; a.x = hrow[kk]; a.y = hrow[kk+1];
      v2f b; b.x = W2f[(size_t)kk*NCOL2 + col]; b.y = W2f[(size_t)(kk+1)*NCOL2 + col];
      acc = __builtin_amdgcn_wmma_f32_16x16x4_f32(false, a, false, b, (short)0, acc, false, false);
    }
    float* wbase = &wsh[wave*16*WSTR2];
    #pragma unroll
    for(int v=0;v<8;++v)
      wbase[(v + 8*kh)*WSTR2 + col] = acc[v] * INV16;
  }
  __syncthreads();
  if (lane < 16){
    int e = blockBase + wave*16 + lane;
    if (e < E){
      const float* wl = &wsh[(wave*16 + lane)*WSTR2];
      int s = esrc[e], d = edst[e];
      const float* zs  = &z[(size_t)s*128];
      const float* eae = &ea[(size_t)e*16];
      float b0 = eae[0];
      float b1[3], b2[5];
      #pragma unroll
      for(int k=0;k<3;++k) b1[k]=eae[1+k];
      #pragma unroll
      for(int j=0;j<5;++j) b2[j]=eae[4+j];
      const float* t4=cg+4*256;  const float* t5=cg+5*256;
      const float* t6=cg+6*256;  const float* t10=cg+10*256;
      float sdot=0.f;
      #pragma unroll
      for(int u=0;u<16;++u) sdot += wl[u]*zs[u];
      float A2[3]={0.f,0.f,0.f}, A3[3]={0.f,0.f,0.f}, A4[3]={0.f,0.f,0.f};
      #pragma unroll
      for(int u=0;u<16;++u){
        float w2c=wl[16+u], w3c=wl[32+u], w4c=wl[48+u];
        #pragma unroll
        for(int i=0;i<3;++i){
          float zo = zs[32+3*u+i];
          A2[i] += w2c*zo;
          A3[i] += w3c*zo;
          A4[i] += w4c*zs[80+3*u+i];
        }
      }
      #pragma unroll
      for(int k=0;k<3;++k){
        float acc = 0.f;
        for(int j=0;j<3;++j) acc += sdot*b1[j]*t4[j*3+k];
        for(int i=0;i<3;++i){
          acc += b0*A2[i]*t5[i*3+k];
          float m=0.f;
          for(int j=0;j<5;++j) m += b2[j]*t6[(i*5+j)*3+k];
          acc += A3[i]*m;
          float nm=0.f;
          for(int j=0;j<3;++j) nm += b1[j]*t10[(i*3+j)*3+k];
          acc += A4[i]*nm;
        }
        atomicAdd(&outp[(size_t)d*3+k], acc*0.125f*INV_NN);  // /sqrt(fan=64)
      }
    }
  }
}

// ---------------------------------------------------------------------------
extern "C" void kernel_launch(void* const* d_in, const int* in_sizes, int n_in,
                              void* d_out, int out_size, void* d_ws, size_t ws_size,
                              hipStream_t stream){
  const float* pos  = (const float*)d_in[0];
  const int*   esrc = (const int*)d_in[1];
  const int*   edst = (const int*)d_in[2];
  const float* W1   = (const float*)d_in[3];
  const float* W2   = (const float*)d_in[4];
  const float* W1f  = (const float*)d_in[5];
  const float* W2f  = (const float*)d_in[6];
  float* outp = (float*)d_out;
  int E = in_sizes[1];
  int N = in_sizes[0]/3;
  (void)n_in; (void)ws_size;

  float* ws = (float*)d_ws;
  size_t off = 0;
  float* cg = ws + off; off += (size_t)13*256;
  float* ea = ws + off; off += (size_t)E*16;
  float* es = ws + off; off += (size_t)E*3;
  float* x0 = ws + off; off += (size_t)N*16;
  float* xg = ws + off; off += (size_t)N*160;
  float* zz = ws + off; off += (size_t)N*128;

  hipMemsetAsync(x0, 0, (size_t)N*16*sizeof(float), stream);
  hipMemsetAsync(xg, 0, (size_t)N*160*sizeof(float), stream);
  hipMemsetAsync(outp, 0, (size_t)out_size*sizeof(float), stream);

  cg_init_kernel<<<1, 32, 0, stream>>>(cg);
  edge_geom_kernel<<<(E+255)/256, 256, 0, stream>>>(pos, esrc, edst, ea, es, x0, E);

  int nb = (E + TILE_E - 1)/TILE_E;
  size_t lds1 = (size_t)(TILE_E*HSTRIDE + 64*WSTR1)*sizeof(float);
  mlp_tp1_kernel<<<nb, BLOCK_T, lds1, stream>>>(es, ea, x0, esrc, edst, W1, W2, cg, xg, E);

  gate_kernel<<<(N*128+255)/256, 256, 0, stream>>>(xg, zz, N);

  size_t lds2 = (size_t)(TILE_E*HSTRIDE + 64*WSTR2)*sizeof(float);
  mlp_tp2_kernel<<<nb, BLOCK_T, lds2, stream>>>(es, ea, zz, esrc, edst, W1f, W2f, cg, outp, E);
}